// dcgcn_34153579938492
// MI455X (gfx1250) — compile-verified
//
#include <hip/hip_runtime.h>

#define U_CNT 100000
#define I_CNT 50000
#define N_CNT 150000
#define N_PAD 150016  // next multiple of 128 >= N_CNT (x buffer padded)
#define D_DIM 64
#define R_CNT 5
#define L_CNT 3
#define E_CNT 500000
#define B_CNT 8192
#define SLOPE 0.01f

typedef __attribute__((ext_vector_type(2))) float v2f;
typedef __attribute__((ext_vector_type(8))) float v8f;

// ---------------------------------------------------------------------------
// x = concat(user_emb, item_emb) (padded rows zeroed); acc = x
// ---------------------------------------------------------------------------
__global__ void dcgcn_init_x(const float* __restrict__ ue,
                             const float* __restrict__ ie,
                             float* __restrict__ x, float* __restrict__ acc) {
  size_t i = (size_t)blockIdx.x * blockDim.x + threadIdx.x;
  if (i >= (size_t)N_PAD * D_DIM) return;
  if (i >= (size_t)N_CNT * D_DIM) {  // zero the pad rows
    x[i] = 0.0f;
    return;
  }
  float v = (i < (size_t)U_CNT * D_DIM) ? ue[i] : ie[i - (size_t)U_CNT * D_DIM];
  x[i] = v;
  acc[i] = v;
}

// ---------------------------------------------------------------------------
// bias1[r][k] = sum_d rating_emb[r+1][d] * aw1[r][64+d][k] + ab1[r][k]
// ---------------------------------------------------------------------------
__global__ void dcgcn_bias(const float* __restrict__ rating_emb,
                           const float* __restrict__ aw1,
                           const float* __restrict__ ab1,
                           float* __restrict__ bias1) {
  int t = blockIdx.x * blockDim.x + threadIdx.x;
  if (t >= R_CNT * D_DIM) return;
  int r = t / D_DIM, k = t % D_DIM;
  const float* re = rating_emb + (size_t)(r + 1) * D_DIM;
  const float* w = aw1 + ((size_t)r * 128 + 64) * 64;  // rows [64,128)
  float s = ab1[(size_t)r * D_DIM + k];
  for (int d = 0; d < D_DIM; ++d) s += re[d] * w[d * 64 + k];
  bias1[t] = s;
}

// ---------------------------------------------------------------------------
// Fused per-rating double GEMM:
//   h2[r] = leaky(x @ aw1[r,:64,:] + bias1[r]) @ aw2[r] + ab2[r]
// One wave = TWO 16x64 row tiles sharing B fragments (weights), f32 WMMA
// 16x16x4. x is padded to N_PAD rows -> no A-side masking needed.
// A/B f32 fragment: idx(lane%16) = M/N, k = 2*(lane/16) + vgpr.
// C/D fragment:     M = vgpr + 8*(lane/16), N = lane%16.
// ---------------------------------------------------------------------------
__global__ __launch_bounds__(128) void dcgcn_fused_gemm(
    const float* __restrict__ x, const float* __restrict__ aw1,
    const float* __restrict__ bias1, const float* __restrict__ aw2,
    const float* __restrict__ ab2, float* __restrict__ h2) {
  __shared__ float w1[64 * 64];
  __shared__ float w2[64 * 64];
  __shared__ float stage[8][16 * 64];  // [wave*2 + tile]

  const int r = blockIdx.y;
  const int tid = threadIdx.x;
  const int wave = tid >> 5;
  const int lane = tid & 31;
  const int lrow = lane & 15;          // M (A/C) or N (B/C) index within tile
  const int khalf = (lane >> 4) * 2;   // k sub-offset for A/B fragments

  const float* gw1 = aw1 + (size_t)r * 128 * 64;  // first 64 rows
  const float* gw2 = aw2 + (size_t)r * 64 * 64;
#pragma unroll
  for (int i = tid * 4; i < 64 * 64; i += 128 * 4) {
    *(float4*)&w1[i] = *(const float4*)&gw1[i];
    *(float4*)&w2[i] = *(const float4*)&gw2[i];
  }
  __syncthreads();

  const int rowbase = blockIdx.x * 128;
  int row0[2];
  row0[0] = rowbase + wave * 16;
  row0[1] = rowbase + 64 + wave * 16;

  const float* xrow[2];
#pragma unroll
  for (int t = 0; t < 2; ++t) xrow[t] = x + (size_t)(row0[t] + lrow) * 64;

  // ---- GEMM1: zero-init accumulators (SRC2 inline 0 on first WMMA)
  v8f c[2][4];
#pragma unroll
  for (int t = 0; t < 2; ++t)
#pragma unroll
    for (int nt = 0; nt < 4; ++nt) c[t][nt] = (v8f){};

#pragma unroll
  for (int kk = 0; kk < 16; ++kk) {
    const int k0 = kk * 4;
    v2f a[2];
#pragma unroll
    for (int t = 0; t < 2; ++t)
      a[t] = *(const v2f*)(xrow[t] + k0 + khalf);  // one b64 load
#pragma unroll
    for (int nt = 0; nt < 4; ++nt) {
      v2f b;
      b.x = w1[(k0 + khalf + 0) * 64 + nt * 16 + lrow];
      b.y = w1[(k0 + khalf + 1) * 64 + nt * 16 + lrow];
      c[0][nt] = __builtin_amdgcn_wmma_f32_16x16x4_f32(false, a[0], false, b,
                                                       (short)0, c[0][nt], false, false);
      c[1][nt] = __builtin_amdgcn_wmma_f32_16x16x4_f32(false, a[1], false, b,
                                                       (short)0, c[1][nt], false, false);
    }
  }

  // ---- bias add + leaky relu; re-fragment C-layout -> A-layout via LDS
#pragma unroll
  for (int nt = 0; nt < 4; ++nt) {
    float bv = bias1[r * 64 + nt * 16 + lrow];
#pragma unroll
    for (int t = 0; t < 2; ++t)
#pragma unroll
      for (int v = 0; v < 8; ++v) {
        float val = c[t][nt][v] + bv;
        val = (val > 0.0f) ? val : SLOPE * val;
        stage[wave * 2 + t][(v + (lane >> 4) * 8) * 64 + nt * 16 + lrow] = val;
      }
  }

  // ---- GEMM2: zero-init accumulators
  v8f c2[2][4];
#pragma unroll
  for (int t = 0; t < 2; ++t)
#pragma unroll
    for (int nt = 0; nt < 4; ++nt) c2[t][nt] = (v8f){};

#pragma unroll
  for (int kk = 0; kk < 16; ++kk) {
    const int k0 = kk * 4;
    v2f a[2];
#pragma unroll
    for (int t = 0; t < 2; ++t)
      a[t] = *(const v2f*)&stage[wave * 2 + t][lrow * 64 + k0 + khalf];  // ds b64
#pragma unroll
    for (int nt = 0; nt < 4; ++nt) {
      v2f b;
      b.x = w2[(k0 + khalf + 0) * 64 + nt * 16 + lrow];
      b.y = w2[(k0 + khalf + 1) * 64 + nt * 16 + lrow];
      c2[0][nt] = __builtin_amdgcn_wmma_f32_16x16x4_f32(false, a[0], false, b,
                                                        (short)0, c2[0][nt], false, false);
      c2[1][nt] = __builtin_amdgcn_wmma_f32_16x16x4_f32(false, a[1], false, b,
                                                        (short)0, c2[1][nt], false, false);
    }
  }

  // ---- add ab2, stage row-major, coalesced float4 stores
  float* h2r = h2 + (size_t)r * N_CNT * 64;
#pragma unroll
  for (int nt = 0; nt < 4; ++nt) {
    float bv = ab2[r * 64 + nt * 16 + lrow];
#pragma unroll
    for (int t = 0; t < 2; ++t)
#pragma unroll
      for (int v = 0; v < 8; ++v)
        stage[wave * 2 + t][(v + (lane >> 4) * 8) * 64 + nt * 16 + lrow] =
            c2[t][nt][v] + bv;
  }

#pragma unroll
  for (int t = 0; t < 2; ++t) {
#pragma unroll
    for (int i = 0; i < 8; ++i) {
      int off = i * 128 + lane * 4;
      int rw = off >> 6;
      int colb = off & 63;
      int grow = row0[t] + rw;
      if (grow < N_CNT) {
        float4 v4 = *(const float4*)&stage[wave * 2 + t][off];
        *(float4*)&h2r[(size_t)grow * 64 + colb] = v4;
      }
    }
  }
}

// ---------------------------------------------------------------------------
// agg[row] += val * h2[r][col]   (summed over all 5 ratings; mean applied later)
// 64 threads per edge -> coalesced 256B gather + contiguous f32 atomics.
// ---------------------------------------------------------------------------
__global__ __launch_bounds__(256) void dcgcn_scatter(
    const int* __restrict__ rows, const int* __restrict__ cols,
    const float* __restrict__ vals, const float* __restrict__ h2,
    float* __restrict__ agg) {
  long long g = (long long)blockIdx.x * 4 + (threadIdx.x >> 6);
  int d = threadIdx.x & 63;
  if (g >= (long long)R_CNT * E_CNT) return;
  int r = (int)(g / E_CNT);
  int row = rows[g];
  int col = cols[g];
  float val = vals[g];
  float hv = h2[((size_t)r * N_CNT + col) * 64 + d];
  atomicAdd(&agg[(size_t)row * 64 + d], val * hv);
}

__global__ void dcgcn_zero(float* __restrict__ p, int n) {
  int i = blockIdx.x * blockDim.x + threadIdx.x;
  if (i < n) p[i] = 0.0f;
}

// x = agg / R; acc += x
__global__ void dcgcn_update(const float* __restrict__ agg,
                             float* __restrict__ x, float* __restrict__ acc,
                             int n) {
  int i = blockIdx.x * blockDim.x + threadIdx.x;
  if (i >= n) return;
  float v = agg[i] * (1.0f / (float)R_CNT);
  x[i] = v;
  acc[i] += v;
}

// ---------------------------------------------------------------------------
// Prediction head: one 64-thread block per (sample, pos/neg).
// light = acc / (L+1); feat = [light_u, light_item]; 128->64->32->1 MLP.
// ---------------------------------------------------------------------------
__global__ __launch_bounds__(64) void dcgcn_mlp(
    const int* __restrict__ users, const int* __restrict__ pos,
    const int* __restrict__ neg, const float* __restrict__ acc,
    const float* __restrict__ pw1, const float* __restrict__ pb1,
    const float* __restrict__ pw2, const float* __restrict__ pb2,
    const float* __restrict__ pw3, const float* __restrict__ pb3,
    float* __restrict__ out) {
  __shared__ float fin[128];
  __shared__ float hh1[64];
  __shared__ float hh2[32];

  int b = blockIdx.x & (B_CNT - 1);
  int isneg = blockIdx.x >> 13;  // B_CNT == 8192 == 2^13
  int t = threadIdx.x;

  int u = users[b];
  int it = isneg ? neg[b] : pos[b];
  const float scale = 1.0f / (float)(L_CNT + 1);
  fin[t] = acc[(size_t)u * 64 + t] * scale;
  fin[64 + t] = acc[((size_t)U_CNT + it) * 64 + t] * scale;
  __syncthreads();

  float s = pb1[t];
  for (int i = 0; i < 128; ++i) s += fin[i] * pw1[i * 64 + t];
  hh1[t] = (s > 0.0f) ? s : SLOPE * s;
  __syncthreads();

  if (t < 32) {
    float s2 = pb2[t];
    for (int i = 0; i < 64; ++i) s2 += hh1[i] * pw2[i * 32 + t];
    hh2[t] = (s2 > 0.0f) ? s2 : SLOPE * s2;
  }
  __syncthreads();

  if (t == 0) {
    float s3 = pb3[0];
    for (int i = 0; i < 32; ++i) s3 += hh2[i] * pw3[i];
    out[(size_t)isneg * B_CNT + b] = s3;
  }
}

// ---------------------------------------------------------------------------
extern "C" void kernel_launch(void* const* d_in, const int* in_sizes, int n_in,
                              void* d_out, int out_size, void* d_ws, size_t ws_size,
                              hipStream_t stream) {
  const int* users = (const int*)d_in[0];
  const int* pos_items = (const int*)d_in[1];
  const int* neg_items = (const int*)d_in[2];
  const float* user_emb = (const float*)d_in[3];
  const float* item_emb = (const float*)d_in[4];
  const float* rating_emb = (const float*)d_in[5];
  const float* aw1 = (const float*)d_in[6];
  const float* ab1 = (const float*)d_in[7];
  const float* aw2 = (const float*)d_in[8];
  const float* ab2 = (const float*)d_in[9];
  const float* pw1 = (const float*)d_in[10];
  const float* pb1 = (const float*)d_in[11];
  const float* pw2 = (const float*)d_in[12];
  const float* pb2 = (const float*)d_in[13];
  const float* pw3 = (const float*)d_in[14];
  const float* pb3 = (const float*)d_in[15];
  const int* edge_rows = (const int*)d_in[16];
  const int* edge_cols = (const int*)d_in[17];
  const float* edge_vals = (const float*)d_in[18];

  // workspace partition (floats)
  const size_t NE = (size_t)N_CNT * D_DIM;        // 9.6M (unpadded arrays)
  const size_t NEP = (size_t)N_PAD * D_DIM;       // padded x
  float* ws = (float*)d_ws;
  float* x = ws;              // N_PAD x 64
  float* acc = x + NEP;       // N x 64
  float* agg = acc + NE;      // N x 64
  float* bias1 = agg + NE;    // R x 64 (+ pad)
  float* h2 = bias1 + 512;    // R x N x 64

  const int nelB = (int)((NE + 255) / 256);
  const int nelBP = (int)((NEP + 255) / 256);

  dcgcn_init_x<<<nelBP, 256, 0, stream>>>(user_emb, item_emb, x, acc);
  dcgcn_bias<<<2, 256, 0, stream>>>(rating_emb, aw1, ab1, bias1);

  dim3 ggrid((N_CNT + 127) / 128, R_CNT);
  const int sgrid = (R_CNT * E_CNT + 3) / 4;
  for (int l = 0; l < L_CNT; ++l) {
    dcgcn_fused_gemm<<<ggrid, 128, 0, stream>>>(x, aw1, bias1, aw2, ab2, h2);
    dcgcn_zero<<<nelB, 256, 0, stream>>>(agg, (int)NE);
    dcgcn_scatter<<<sgrid, 256, 0, stream>>>(edge_rows, edge_cols, edge_vals, h2, agg);
    dcgcn_update<<<nelB, 256, 0, stream>>>(agg, x, acc, (int)NE);
  }

  dcgcn_mlp<<<2 * B_CNT, 64, 0, stream>>>(users, pos_items, neg_items, acc,
                                          pw1, pb1, pw2, pb2, pw3, pb3,
                                          (float*)d_out);
}